// GCN_7112465842754
// MI455X (gfx1250) — compile-verified
//
#include <hip/hip_runtime.h>
#include <math.h>

// GCN forward for MI455X (gfx1250, wave32).
// Dense GEMMs use V_WMMA_F32_16X16X4_F32 (fp32 WMMA, fp32 accumulate — matches
// reference numerics). SpMM is CSR-style over the sorted COO rows; the gathered
// dense operands (102.4 MB / 6.4 MB) are L2-resident (192 MB L2).

#define NFEAT 512
#define NHID  256
#define NCLASS 16

typedef __attribute__((ext_vector_type(2))) float v2f;
typedef __attribute__((ext_vector_type(8))) float v8f;

__device__ __forceinline__ v8f wmma_f32(v2f a, v2f b, v8f c) {
  // 8 args: (neg_a, A, neg_b, B, c_mod, C, reuse_a, reuse_b)
  return __builtin_amdgcn_wmma_f32_16x16x4_f32(false, a, false, b, (short)0, c,
                                               false, false);
}

// ---------------------------------------------------------------- row pointers
__global__ void build_rowptr(const int* __restrict__ row, int E, int N,
                             int* __restrict__ ptr) {
  int r = blockIdx.x * blockDim.x + threadIdx.x;
  if (r > N) return;
  int lo = 0, hi = E;                       // lower_bound(row, r)
  while (lo < hi) {
    int mid = (lo + hi) >> 1;
    if (row[mid] < r) lo = mid + 1; else hi = mid;
  }
  ptr[r] = lo;
}

// --------------------------------------------------- GEMM1: [N,512]@[512,256]
// Block = 256 threads (8 waves). Block computes 16 rows x 256 cols.
// A-tile (16x512) staged in LDS, padded to stride 516 floats: lane L reads
// row L%16 at dword k+2*(L/16) -> banks {4m+k .. 4m+k+3} cover all 64 banks.
__global__ __launch_bounds__(256) void gemm1_wmma(const float* __restrict__ X,
                                                  const float* __restrict__ W1,
                                                  float* __restrict__ H0) {
  __shared__ float lds[16 * 516];
  const int tid = threadIdx.x;
  const int row0 = blockIdx.x * 16;

  for (int i = tid; i < 16 * 512; i += 256) {
    int rr = i >> 9, kk = i & 511;
    lds[rr * 516 + kk] = X[(size_t)(row0 + rr) * NFEAT + kk];
  }
  __syncthreads();

  const int wave = tid >> 5, lane = tid & 31;
  const int ln = lane & 15;      // M for A / N for B,C,D
  const int kh = lane >> 4;      // K-half selector
  const int n0 = wave * 32;      // two adjacent 16-wide column tiles

  v8f acc0 = {}; v8f acc1 = {};
  for (int k = 0; k < NFEAT; k += 4) {
    v2f a;
    a.x = lds[ln * 516 + k + 2 * kh];
    a.y = lds[ln * 516 + k + 2 * kh + 1];
    const float* w0 = W1 + (size_t)(k + 2 * kh) * NHID;  // K row, next row +NHID
    v2f b0, b1v;
    b0.x  = w0[n0 + ln];        b0.y  = w0[NHID + n0 + ln];
    b1v.x = w0[n0 + 16 + ln];   b1v.y = w0[NHID + n0 + 16 + ln];
    acc0 = wmma_f32(a, b0, acc0);
    acc1 = wmma_f32(a, b1v, acc1);
  }
#pragma unroll
  for (int v = 0; v < 8; ++v) {        // D layout: M = v + 8*kh, N = ln
    int r = row0 + v + 8 * kh;
    H0[(size_t)r * NHID + n0 + ln]      = acc0[v];
    H0[(size_t)r * NHID + n0 + 16 + ln] = acc1[v];
  }
}

// ---------------------------- SpMM over hidden dim + bias + ReLU (D = 256)
// One block per node, thread = channel. Gathered rows (1 KB) are L2-resident.
__global__ __launch_bounds__(256) void spmm_hid(const int* __restrict__ ptr,
                                                const int* __restrict__ col,
                                                const float* __restrict__ vals,
                                                const float* __restrict__ H0,
                                                const float* __restrict__ b1,
                                                float* __restrict__ H) {
  const int r = blockIdx.x;
  const int c = threadIdx.x;
  const int e0 = ptr[r], e1 = ptr[r + 1];
  float acc = 0.f;
  for (int e = e0; e < e1; ++e)
    acc = fmaf(vals[e], H0[(size_t)col[e] * NHID + c], acc);
  float h = acc + b1[c];
  H[(size_t)r * NHID + c] = h > 0.f ? h : 0.f;
}

// --------------------------------------------------- GEMM2: [N,256]@[256,16]
__global__ __launch_bounds__(256) void gemm2_wmma(const float* __restrict__ H,
                                                  const float* __restrict__ W2,
                                                  float* __restrict__ H2,
                                                  int nRows) {
  const int tid = threadIdx.x;
  const int wave = tid >> 5, lane = tid & 31;
  const int row0 = (blockIdx.x * 8 + wave) * 16;
  if (row0 >= nRows) return;               // wave-uniform: EXEC stays all-ones
  const int ln = lane & 15, kh = lane >> 4;

  v8f acc = {};
  for (int k = 0; k < NHID; k += 4) {
    v2f a, b;
    const float* ha = H + (size_t)(row0 + ln) * NHID + k + 2 * kh;
    a.x = ha[0]; a.y = ha[1];
    const float* wb = W2 + (size_t)(k + 2 * kh) * NCLASS + ln;
    b.x = wb[0]; b.y = wb[NCLASS];
    acc = wmma_f32(a, b, acc);
  }
#pragma unroll
  for (int v = 0; v < 8; ++v)
    H2[(size_t)(row0 + v + 8 * kh) * NCLASS + ln] = acc[v];
}

// -------- Final: SpMM (D=16) for both branches + b2 + fused log-softmax(32)
// One wave per node: lanes 0-15 = branch x, lanes 16-31 = branch y.
__global__ __launch_bounds__(256) void spmm_out_softmax(
    const int* __restrict__ ptr, const int* __restrict__ col,
    const float* __restrict__ vals, const float* __restrict__ H2x,
    const float* __restrict__ H2y, const float* __restrict__ b2,
    float* __restrict__ out, int N) {
  const int wave = threadIdx.x >> 5;
  const int lane = threadIdx.x & 31;
  const int r = blockIdx.x * 8 + wave;
  if (r >= N) return;                      // wave-uniform
  const int c = lane & 15;
  const float* P = (lane < 16) ? H2x : H2y;

  const int e0 = ptr[r], e1 = ptr[r + 1];
  float acc = 0.f;
  for (int e = e0; e < e1; ++e)
    acc = fmaf(vals[e], P[(size_t)col[e] * NCLASS + c], acc);
  float v = acc + b2[c];

  float m = v;                             // max over all 32 lanes
#pragma unroll
  for (int off = 16; off; off >>= 1) m = fmaxf(m, __shfl_xor(m, off, 32));
  float s = __expf(v - m);                 // sum of exp
#pragma unroll
  for (int off = 16; off; off >>= 1) s += __shfl_xor(s, off, 32);

  out[(size_t)r * 32 + lane] = v - m - __logf(s);
}

// ----------------------------------------------------------------- launcher
extern "C" void kernel_launch(void* const* d_in, const int* in_sizes, int n_in,
                              void* d_out, int out_size, void* d_ws,
                              size_t ws_size, hipStream_t stream) {
  const float* x    = (const float*)d_in[0];
  const float* y    = (const float*)d_in[1];
  const int*   row  = (const int*)  d_in[2];
  const int*   col  = (const int*)  d_in[3];
  const float* vals = (const float*)d_in[4];
  const float* W1   = (const float*)d_in[5];
  const float* b1   = (const float*)d_in[6];
  const float* W2   = (const float*)d_in[7];
  const float* b2   = (const float*)d_in[8];
  float* out = (float*)d_out;

  const int N = in_sizes[0] / NFEAT;   // 100000
  const int E = in_sizes[2];           // 3200000

  // Workspace layout (bytes): ptr | H0 (102.4MB) | H (102.4MB) | H2x | H2y
  char* w = (char*)d_ws;
  int*   ptr = (int*)w;
  float* H0  = (float*)(w + (4ull   << 20));
  float* H   = (float*)(w + (108ull << 20));
  float* H2x = (float*)(w + (212ull << 20));
  float* H2y = (float*)(w + (220ull << 20));

  build_rowptr<<<(N + 256) / 256, 256, 0, stream>>>(row, E, N, ptr);

  const float* feats[2] = {x, y};
  float* h2s[2] = {H2x, H2y};
  for (int b = 0; b < 2; ++b) {
    gemm1_wmma<<<N / 16, 256, 0, stream>>>(feats[b], W1, H0);
    spmm_hid<<<N, 256, 0, stream>>>(ptr, col, vals, H0, b1, H);
    gemm2_wmma<<<(N / 16 + 7) / 8, 256, 0, stream>>>(H, W2, h2s[b], N);
  }
  spmm_out_softmax<<<(N + 7) / 8, 256, 0, stream>>>(ptr, col, vals, H2x, H2y,
                                                    b2, out, N);
}